// SparseMoeWrapper_18279380812579
// MI455X (gfx1250) — compile-verified
//
#include <hip/hip_runtime.h>
#include <hip/hip_bf16.h>
#include <math.h>

// Problem constants (B=2, S=1024 -> T=2048)
#define T_TOK 2048
#define H_DIM 2048
#define F_DIM 7168
#define E_NUM 8
#define KC    32          // K-chunk staged in LDS (per buffer)
#define PAD   4           // row pad: keeps rows 16B-aligned + bank-conflict free
#define MT    64          // token tile per block (4 WMMA M-subtiles)

typedef __attribute__((ext_vector_type(2))) float v2f;
typedef __attribute__((ext_vector_type(8))) float v8f;
typedef __attribute__((ext_vector_type(4))) int   v4i;

#define GLOBAL_AS __attribute__((address_space(1)))
#define LDS_AS    __attribute__((address_space(3)))

#if defined(__has_builtin)
#if __has_builtin(__builtin_amdgcn_global_load_async_to_lds_b128)
#define HAVE_ASYNC_LDS 1
#endif
#if __has_builtin(__builtin_amdgcn_s_wait_asynccnt)
#define HAVE_WAIT_ASYNC 1
#endif
#endif

__device__ __forceinline__ void async_copy16(const float* gp, float* lp) {
#if defined(HAVE_ASYNC_LDS)
  float* g = const_cast<float*>(gp);
  __builtin_amdgcn_global_load_async_to_lds_b128(
      (GLOBAL_AS v4i*)g, (LDS_AS v4i*)lp, 0, 0);
#else
  lp[0] = gp[0]; lp[1] = gp[1]; lp[2] = gp[2]; lp[3] = gp[3];
#endif
}

__device__ __forceinline__ void wait_async_lds() {
#if defined(HAVE_ASYNC_LDS)
#if defined(HAVE_WAIT_ASYNC)
  __builtin_amdgcn_s_wait_asynccnt(0);
#else
  asm volatile("s_wait_asynccnt 0x0" ::: "memory");
#endif
#endif
}

// ---------------------------------------------------------------------------
// Kernel 0: zero output accumulator region + expert counters
// ---------------------------------------------------------------------------
__global__ void zero_kernel(float* __restrict__ out, int n_out,
                            int* __restrict__ cnts) {
  int i = blockIdx.x * blockDim.x + threadIdx.x;
  if (i < n_out) out[i] = 0.0f;
  if (i < E_NUM) cnts[i] = 0;
}

// ---------------------------------------------------------------------------
// Kernel 1: router. One block (8 waves) per token; wave e computes logit[e].
// ---------------------------------------------------------------------------
__global__ __launch_bounds__(256)
void router_kernel(const float* __restrict__ x,
                   const float* __restrict__ Wg,
                   float* __restrict__ logits_out,   // [T,E]
                   float* __restrict__ combine,      // [T,E]
                   int* __restrict__ cnts,           // [E]
                   int* __restrict__ lists) {        // [E][T], entry = t*2+k
  const int t    = blockIdx.x;
  const int e    = threadIdx.x >> 5;
  const int lane = threadIdx.x & 31;
  __shared__ float s_logit[E_NUM];

  const float* xr = x  + (size_t)t * H_DIM;
  const float* wr = Wg + (size_t)e * H_DIM;
  float p = 0.f;
  for (int h = lane; h < H_DIM; h += 32) p += xr[h] * wr[h];
  for (int off = 16; off > 0; off >>= 1) p += __shfl_xor(p, off, 32);
  if (lane == 0) {
    s_logit[e] = p;
    logits_out[t * E_NUM + e] = p;
  }
  __syncthreads();

  if (threadIdx.x == 0) {
    float mx = s_logit[0];
    for (int i = 1; i < E_NUM; i++) mx = fmaxf(mx, s_logit[i]);
    float pe[E_NUM]; float sum = 0.f;
    for (int i = 0; i < E_NUM; i++) { pe[i] = __expf(s_logit[i] - mx); sum += pe[i]; }
    float inv = 1.f / sum;
    for (int i = 0; i < E_NUM; i++) pe[i] *= inv;
    int i0 = 0;
    for (int i = 1; i < E_NUM; i++) if (pe[i] > pe[i0]) i0 = i;
    int i1 = -1;
    for (int i = 0; i < E_NUM; i++) {
      if (i == i0) continue;
      if (i1 < 0 || pe[i] > pe[i1]) i1 = i;
    }
    float w0 = pe[i0], w1 = pe[i1];
    float rs = 1.f / (w0 + w1);
    w0 *= rs; w1 *= rs;
    for (int i = 0; i < E_NUM; i++)
      combine[t * E_NUM + i] = (i == i0) ? w0 : ((i == i1) ? w1 : 0.f);
    int s0 = atomicAdd(&cnts[i0], 1); lists[i0 * T_TOK + s0] = t * 2 + 0;
    int s1 = atomicAdd(&cnts[i1], 1); lists[i1 * T_TOK + s1] = t * 2 + 1;
  }
}

// ---------------------------------------------------------------------------
// Kernel 2: h = silu(x @ w1[e]^T) * (x @ w3[e]^T) for routed pairs.
// Block = 128 threads (4 waves). Tile: 64 tokens x 64 F. Double-buffered
// async global->LDS staging over H in KC chunks; WMMA f32 16x16x4 core.
// ---------------------------------------------------------------------------
__global__ __launch_bounds__(128)
void ffn_h_kernel(const float* __restrict__ x,
                  const float* __restrict__ w1,
                  const float* __restrict__ w3,
                  const int* __restrict__ cnts,
                  const int* __restrict__ lists,
                  float* __restrict__ hbuf) {        // [2T][F]
  const int e     = blockIdx.z;
  const int cnt   = cnts[e];
  const int tile0 = blockIdx.y * MT;
  if (tile0 >= cnt) return;
  const int fblk = blockIdx.x * 64;
  const int tid  = threadIdx.x;
  const int wave = tid >> 5;
  const int lane = tid & 31;

  __shared__ int   s_ent[MT];
  __shared__ float xs [2][MT][KC + PAD];
  __shared__ float b1s[2][64][KC + PAD];
  __shared__ float b3s[2][64][KC + PAD];

  if (tid < MT) {
    int idx = tile0 + tid;
    s_ent[tid] = (idx < cnt) ? lists[e * T_TOK + idx] : -1;
  }
  __syncthreads();

  const size_t wbase = (size_t)e * F_DIM * H_DIM + (size_t)fblk * H_DIM;

  // stage chunk `kk` into buffer `b` (16B segments; async fire-and-forget)
  auto stage = [&](int b, int kk) {
    const int SEG = KC / 4;  // 16B segments per row
    for (int s = tid; s < MT * SEG; s += 128) {
      int r = s / SEG, c4 = s % SEG;
      int ent = s_ent[r];
      int tok = (ent < 0) ? 0 : (ent >> 1);
      async_copy16(&x[(size_t)tok * H_DIM + kk + c4 * 4], &xs[b][r][c4 * 4]);
    }
    for (int s = tid; s < 64 * SEG; s += 128) {
      int r = s / SEG, c4 = s % SEG;
      size_t g = wbase + (size_t)r * H_DIM + kk + c4 * 4;
      async_copy16(&w1[g], &b1s[b][r][c4 * 4]);
      async_copy16(&w3[g], &b3s[b][r][c4 * 4]);
    }
  };

  v8f c1[4] = {}, c3[4] = {};
  const int m  = lane & 15;          // row within 16-token subtile
  const int wn = wave * 16 + (lane & 15);
  const int kb = (lane >> 4) * 2;    // K sub-offset per ISA lane layout

  const int NCH = H_DIM / KC;
  stage(0, 0);
  for (int i = 0; i < NCH; i++) {
    wait_async_lds();
    __syncthreads();                 // chunk i landed; prev buffer free
    if (i + 1 < NCH) stage((i + 1) & 1, (i + 1) * KC);

    const int b = i & 1;
    const float* bp1 = &b1s[b][wn][0];
    const float* bp3 = &b3s[b][wn][0];
#pragma unroll
    for (int k4 = 0; k4 < KC; k4 += 4) {
      v2f b1v, b3v;
      b1v.x = bp1[k4 + kb]; b1v.y = bp1[k4 + kb + 1];
      b3v.x = bp3[k4 + kb]; b3v.y = bp3[k4 + kb + 1];
#pragma unroll
      for (int ms = 0; ms < 4; ms++) {
        const float* ap = &xs[b][ms * 16 + m][0];
        v2f a; a.x = ap[k4 + kb]; a.y = ap[k4 + kb + 1];
        c1[ms] = __builtin_amdgcn_wmma_f32_16x16x4_f32(
            false, a, false, b1v, (short)0, c1[ms], false, false);
        c3[ms] = __builtin_amdgcn_wmma_f32_16x16x4_f32(
            false, a, false, b3v, (short)0, c3[ms], false, false);
      }
    }
  }

  // h = silu(h1) * h3, scatter to hbuf rows (pair-indexed: ent = t*2+k)
  const int n = lane & 15;
  for (int ms = 0; ms < 4; ms++) {
    for (int r = 0; r < 8; r++) {
      int mm  = ms * 16 + r + ((lane >> 4) * 8);
      int ent = s_ent[mm];
      if (ent < 0) continue;
      float g = c1[ms][r];
      float hv = (g / (1.0f + __expf(-g))) * c3[ms][r];
      hbuf[(size_t)ent * F_DIM + fblk + wave * 16 + n] = hv;
    }
  }
}

// ---------------------------------------------------------------------------
// Kernel 3: out[t] += combine[t,e] * (h @ w2[e]^T). 64x64 tile, K over F,
// double-buffered async staging. Exactly 2 commutative fp32 atomics/elem.
// ---------------------------------------------------------------------------
__global__ __launch_bounds__(128)
void ffn_out_kernel(const float* __restrict__ hbuf,
                    const float* __restrict__ w2,
                    const float* __restrict__ combine,
                    const int* __restrict__ cnts,
                    const int* __restrict__ lists,
                    float* __restrict__ out) {
  const int e     = blockIdx.z;
  const int cnt   = cnts[e];
  const int tile0 = blockIdx.y * MT;
  if (tile0 >= cnt) return;
  const int hblk = blockIdx.x * 64;
  const int tid  = threadIdx.x;
  const int wave = tid >> 5;
  const int lane = tid & 31;

  __shared__ int   s_ent[MT];
  __shared__ float as [2][MT][KC + PAD];
  __shared__ float b2s[2][64][KC + PAD];

  if (tid < MT) {
    int idx = tile0 + tid;
    s_ent[tid] = (idx < cnt) ? lists[e * T_TOK + idx] : -1;
  }
  __syncthreads();

  const size_t wbase = (size_t)e * H_DIM * F_DIM + (size_t)hblk * F_DIM;

  auto stage = [&](int b, int kk) {
    const int SEG = KC / 4;
    for (int s = tid; s < MT * SEG; s += 128) {
      int r = s / SEG, c4 = s % SEG;
      int ent = s_ent[r];
      int row = (ent < 0) ? 0 : ent;
      async_copy16(&hbuf[(size_t)row * F_DIM + kk + c4 * 4], &as[b][r][c4 * 4]);
    }
    for (int s = tid; s < 64 * SEG; s += 128) {
      int r = s / SEG, c4 = s % SEG;
      async_copy16(&w2[wbase + (size_t)r * F_DIM + kk + c4 * 4],
                   &b2s[b][r][c4 * 4]);
    }
  };

  v8f c[4] = {};
  const int m  = lane & 15;
  const int wn = wave * 16 + (lane & 15);
  const int kb = (lane >> 4) * 2;

  const int NCH = F_DIM / KC;
  stage(0, 0);
  for (int i = 0; i < NCH; i++) {
    wait_async_lds();
    __syncthreads();
    if (i + 1 < NCH) stage((i + 1) & 1, (i + 1) * KC);

    const int b = i & 1;
    const float* bp = &b2s[b][wn][0];
#pragma unroll
    for (int k4 = 0; k4 < KC; k4 += 4) {
      v2f bv; bv.x = bp[k4 + kb]; bv.y = bp[k4 + kb + 1];
#pragma unroll
      for (int ms = 0; ms < 4; ms++) {
        const float* ap = &as[b][ms * 16 + m][0];
        v2f a; a.x = ap[k4 + kb]; a.y = ap[k4 + kb + 1];
        c[ms] = __builtin_amdgcn_wmma_f32_16x16x4_f32(
            false, a, false, bv, (short)0, c[ms], false, false);
      }
    }
  }

  const int n = lane & 15;
  for (int ms = 0; ms < 4; ms++) {
    for (int r = 0; r < 8; r++) {
      int mm  = ms * 16 + r + ((lane >> 4) * 8);
      int ent = s_ent[mm];
      if (ent < 0) continue;
      int tok = ent >> 1;
      float w = combine[tok * E_NUM + e];
      atomicAdd(&out[(size_t)tok * H_DIM + hblk + wave * 16 + n], c[ms][r] * w);
    }
  }
}

// ---------------------------------------------------------------------------
// Host launcher.
// d_in: [hidden_states, Wg, w1, w3, w2] (all float32)
// d_out: out [T*H] floats, then router_logits [T*E] floats
// ws layout: combine (64KB) | cnts (256B) | lists (64KB) | hbuf (117MB)
// ---------------------------------------------------------------------------
extern "C" void kernel_launch(void* const* d_in, const int* in_sizes, int n_in,
                              void* d_out, int out_size, void* d_ws, size_t ws_size,
                              hipStream_t stream) {
  const float* x  = (const float*)d_in[0];
  const float* Wg = (const float*)d_in[1];
  const float* w1 = (const float*)d_in[2];
  const float* w3 = (const float*)d_in[3];
  const float* w2 = (const float*)d_in[4];

  float* out    = (float*)d_out;                       // [T*H]
  float* logits = out + (size_t)T_TOK * H_DIM;         // [T*E]

  unsigned char* ws = (unsigned char*)d_ws;
  float* combine = (float*)ws;                                         // T*E f32
  int*   cnts    = (int*)(ws + (size_t)T_TOK * E_NUM * 4);             // E ints
  int*   lists   = (int*)(ws + (size_t)T_TOK * E_NUM * 4 + 256);       // E*T ints
  float* hbuf    = (float*)(ws + (size_t)T_TOK * E_NUM * 4 + 256
                               + (size_t)E_NUM * T_TOK * 4);           // 2T*F f32

  const int n_out = T_TOK * H_DIM;
  zero_kernel<<<(n_out + 255) / 256, 256, 0, stream>>>(out, n_out, cnts);

  router_kernel<<<T_TOK, 256, 0, stream>>>(x, Wg, logits, combine, cnts, lists);

  dim3 g2(F_DIM / 64, T_TOK / MT, E_NUM);
  ffn_h_kernel<<<g2, 128, 0, stream>>>(x, w1, w3, cnts, lists, hbuf);

  dim3 g3(H_DIM / 64, T_TOK / MT, E_NUM);
  ffn_out_kernel<<<g3, 128, 0, stream>>>(hbuf, w2, combine, cnts, lists, out);
}